// Predictor_11922829214129
// MI455X (gfx1250) — compile-verified
//
#include <hip/hip_runtime.h>
#include <hip/hip_bf16.h>

// Problem constants (from reference setup_inputs)
#define BB 256
#define CC 64
#define DD 256
#define HH 4096
#define OO 256
#define EE 4
#define EPSV 1e-5f

typedef float v2f __attribute__((ext_vector_type(2)));
typedef float v8f __attribute__((ext_vector_type(8)));

// D = A(16x4 f32) * B(4x16 f32) + C(16x16 f32)
__device__ __forceinline__ v8f wmma_f32(v2f a, v2f b, v8f c) {
  return __builtin_amdgcn_wmma_f32_16x16x4_f32(
      /*neg_a=*/false, a, /*neg_b=*/false, b,
      /*c_mod=*/(short)0, c, /*reuse_a=*/false, /*reuse_b=*/false);
}

// Async global->LDS copy of 16 bytes per lane (CDNA5 path, ASYNCcnt-tracked,
// bypasses VGPRs). Low 32 bits of a generic LDS pointer are the LDS address.
__device__ __forceinline__ void async_b128(const void* g, void* l) {
  asm volatile("global_load_async_to_lds_b128 %0, %1, off"
               :
               : "v"((unsigned int)(unsigned long long)(size_t)l), "v"(g)
               : "memory");
}
#define WAIT_ASYNC(n) asm volatile("s_wait_asynccnt " #n ::: "memory")

// ---------------------------------------------------------------------------
// Phase 1: h[b,c,:] = x[b,c,:] @ W1[e_c] + b1[e_c]
// Block tile 128(B) x 128(H) per channel; double-buffered LDS with async
// global->LDS staging. 8 waves: 4 along M (32 rows each) x 2 along N (64 cols);
// wave computes 2x4 WMMA subtiles (B frag reused 2x, A frag 4x).
// Also emits per-block column partials (sum, sum^2) for BatchNorm.
// grid = (H/128=32, B/128=2, C=64), block = 256
// ---------------------------------------------------------------------------
__global__ __launch_bounds__(256) void moe_gemm1_kernel(
    const float* __restrict__ x,     // [B,C,D]
    const float* __restrict__ W1,    // [E,D,H]
    const float* __restrict__ b1,    // [E,H]
    const int*   __restrict__ pidx,  // [C]
    float* __restrict__ h,           // [B,C,H]
    float* __restrict__ part1,       // [2,C,H]
    float* __restrict__ part2)       // [2,C,H]
{
  __shared__ float As[2][128][20];   // A tiles, K-contig rows, 16B-aligned pad
  __shared__ float Bs[2][16][136];   // B tiles, k-major, bank-spread pad
  __shared__ float cs1[128];
  __shared__ float cs2[128];

  const int c     = blockIdx.z;
  const int hbase = blockIdx.x * 128;
  const int bbase = blockIdx.y * 128;
  const int e     = pidx[c];

  const int t    = threadIdx.x;
  const int lane = t & 31;
  const int wave = t >> 5;
  const int wm   = wave & 3;   // 32 M-rows per wave
  const int wn   = wave >> 2;  // 64 N-cols per wave
  const int lh   = lane >> 4;
  const int ll   = lane & 15;

  if (t < 128) { cs1[t] = 0.0f; cs2[t] = 0.0f; }

  v8f acc[2][4];
#pragma unroll
  for (int mt = 0; mt < 2; ++mt)
#pragma unroll
    for (int nt = 0; nt < 4; ++nt) acc[mt][nt] = (v8f)(0.0f);

  // staging maps: A: thread -> (row = t>>1, 8 k's at (t&1)*8)
  //               B: thread -> (k-row = t>>4, 8 cols at (t&15)*8)
  const int arow = t >> 1;
  const int akk  = (t & 1) * 8;
  const int bkr  = t >> 4;
  const int bcol = (t & 15) * 8;

  const float* xrow = &x[((size_t)(bbase + arow) * CC + c) * DD];
  const float* wrow = &W1[((size_t)e * DD + bkr) * HH + hbase + bcol];

  auto issue_stage = [&](int k0, int buf) {
    async_b128(xrow + k0 + akk,             &As[buf][arow][akk]);
    async_b128(xrow + k0 + akk + 4,         &As[buf][arow][akk + 4]);
    async_b128(wrow + (size_t)k0 * HH,      &Bs[buf][bkr][bcol]);
    async_b128(wrow + (size_t)k0 * HH + 4,  &Bs[buf][bkr][bcol + 4]);
  };

  issue_stage(0, 0);
  int buf = 0;
  for (int k0 = 0; k0 < DD; k0 += 16) {
    if (k0 + 16 < DD) {
      issue_stage(k0 + 16, buf ^ 1);  // prefetch next stage (4 async ops)
      WAIT_ASYNC(4);                  // current stage's 4 ops complete
    } else {
      WAIT_ASYNC(0);
    }
    __syncthreads();

#pragma unroll
    for (int ks = 0; ks < 4; ++ks) {
      const int kf = ks * 4 + 2 * lh;
      const v2f a0 = *(const v2f*)&As[buf][wm * 32 + ll][kf];
      const v2f a1 = *(const v2f*)&As[buf][wm * 32 + 16 + ll][kf];
#pragma unroll
      for (int nt = 0; nt < 4; ++nt) {
        const int coln = wn * 64 + nt * 16 + ll;
        v2f b;
        b[0] = Bs[buf][kf][coln];
        b[1] = Bs[buf][kf + 1][coln];
        acc[0][nt] = wmma_f32(a0, b, acc[0][nt]);
        acc[1][nt] = wmma_f32(a1, b, acc[1][nt]);
      }
    }
    __syncthreads();
    buf ^= 1;
  }

  // bias add + per-column stats + store h
#pragma unroll
  for (int nt = 0; nt < 4; ++nt) {
    const int coln = wn * 64 + nt * 16 + ll;
    const float bias = b1[(size_t)e * HH + hbase + coln];
    float s = 0.0f, q = 0.0f;
#pragma unroll
    for (int mt = 0; mt < 2; ++mt)
#pragma unroll
      for (int j = 0; j < 8; ++j) {
        const float v = acc[mt][nt][j] + bias;
        acc[mt][nt][j] = v;
        s += v; q += v * v;
      }
    atomicAdd(&cs1[coln], s);
    atomicAdd(&cs2[coln], q);
    const int colg = hbase + coln;
#pragma unroll
    for (int mt = 0; mt < 2; ++mt)
#pragma unroll
      for (int j = 0; j < 8; ++j) {
        const int row = bbase + wm * 32 + mt * 16 + j + 8 * lh;
        h[((size_t)row * CC + c) * HH + colg] = acc[mt][nt][j];
      }
  }
  __syncthreads();
  if (t < 128) {
    const size_t o = ((size_t)blockIdx.y * CC + c) * HH + hbase + t;
    part1[o] = cs1[t];
    part2[o] = cs2[t];
  }
}

// ---------------------------------------------------------------------------
// Phase 1b: deterministic reduce of block partials into per-expert s1/s2.
// grid = E*H/256 = 64, block = 256. thread -> (e, hcol)
// ---------------------------------------------------------------------------
__global__ __launch_bounds__(256) void moe_stats_kernel(
    const float* __restrict__ part1, const float* __restrict__ part2,
    const int* __restrict__ pidx,
    float* __restrict__ s1, float* __restrict__ s2)
{
  const int gid = blockIdx.x * blockDim.x + threadIdx.x;   // e*H + hcol
  const int e  = gid >> 12;
  const int hc = gid & (HH - 1);
  float a = 0.0f, b = 0.0f;
  for (int bt = 0; bt < 2; ++bt)
    for (int c = 0; c < CC; ++c)
      if (pidx[c] == e) {
        const size_t o = ((size_t)bt * CC + c) * HH + hc;
        a += part1[o];
        b += part2[o];
      }
  s1[gid] = a;
  s2[gid] = b;
}

// ---------------------------------------------------------------------------
// Phase 2: fold BN (train-mode, biased var) into per-(e,h) scale/shift.
// ---------------------------------------------------------------------------
__global__ __launch_bounds__(256) void moe_bnparam_kernel(
    const float* __restrict__ s1, const float* __restrict__ s2,
    const int* __restrict__ pidx,
    const float* __restrict__ gamma, const float* __restrict__ beta,
    float* __restrict__ scale, float* __restrict__ shift)
{
  const int gid = blockIdx.x * blockDim.x + threadIdx.x;   // e*H + hcol
  const int e = gid >> 12;
  int nch = 0;
  for (int c = 0; c < CC; ++c) nch += (pidx[c] == e) ? 1 : 0;
  const float cnt  = fmaxf((float)(nch * BB), 1.0f);
  const float mean = s1[gid] / cnt;
  const float var  = s2[gid] / cnt - mean * mean;
  const float inv  = rsqrtf(var + EPSV);
  const float sc   = inv * gamma[gid];
  scale[gid] = sc;
  shift[gid] = beta[gid] - mean * sc;
}

// ---------------------------------------------------------------------------
// Phase 3: out[b,c,:] = relu(h*scale+shift) @ W2[e_c] + b2[e_c]
// A tile staged via VGPR path (fused BN+ReLU, indexed by K=hcol), B tile via
// async global->LDS; both double-buffered. Same 128x128 block / 32x64 wave
// tiling as phase 1.
// grid = (O/128=2, B/128=2, C=64), block = 256
// ---------------------------------------------------------------------------
__global__ __launch_bounds__(256) void moe_gemm2_kernel(
    const float* __restrict__ h,     // [B,C,H]
    const float* __restrict__ W2,    // [E,H,O]
    const float* __restrict__ b2,    // [E,O]
    const int*   __restrict__ pidx,  // [C]
    const float* __restrict__ scale, // [E,H]
    const float* __restrict__ shift, // [E,H]
    float* __restrict__ out)         // [B,C,O]
{
  __shared__ float As[2][128][20];
  __shared__ float Bs[2][16][136];

  const int c     = blockIdx.z;
  const int obase = blockIdx.x * 128;
  const int bbase = blockIdx.y * 128;
  const int e     = pidx[c];

  const int t    = threadIdx.x;
  const int lane = t & 31;
  const int wave = t >> 5;
  const int wm   = wave & 3;
  const int wn   = wave >> 2;
  const int lh   = lane >> 4;
  const int ll   = lane & 15;

  v8f acc[2][4];
#pragma unroll
  for (int mt = 0; mt < 2; ++mt)
#pragma unroll
    for (int nt = 0; nt < 4; ++nt) acc[mt][nt] = (v8f)(0.0f);

  const int arow = t >> 1;
  const int akk  = (t & 1) * 8;
  const int bkr  = t >> 4;
  const int bcol = (t & 15) * 8;

  const float* hrow  = &h[((size_t)(bbase + arow) * CC + c) * HH];
  const float* w2row = &W2[((size_t)e * HH + bkr) * OO + obase + bcol];
  const float* scl   = &scale[(size_t)e * HH];
  const float* shf   = &shift[(size_t)e * HH];

  auto issue_b = [&](int k0, int buf) {
    async_b128(w2row + (size_t)k0 * OO,     &Bs[buf][bkr][bcol]);
    async_b128(w2row + (size_t)k0 * OO + 4, &Bs[buf][bkr][bcol + 4]);
  };
  auto stage_a = [&](int k0, int buf) {
#pragma unroll
    for (int p = 0; p < 2; ++p) {
      const int kk = k0 + akk + p * 4;
      const float4 hv = *(const float4*)&hrow[kk];
      const float4 sv = *(const float4*)&scl[kk];
      const float4 fv = *(const float4*)&shf[kk];
      float4 r;
      r.x = fmaxf(fmaf(hv.x, sv.x, fv.x), 0.0f);
      r.y = fmaxf(fmaf(hv.y, sv.y, fv.y), 0.0f);
      r.z = fmaxf(fmaf(hv.z, sv.z, fv.z), 0.0f);
      r.w = fmaxf(fmaf(hv.w, sv.w, fv.w), 0.0f);
      *(float4*)&As[buf][arow][akk + p * 4] = r;
    }
  };

  issue_b(0, 0);
  stage_a(0, 0);
  int buf = 0;
  for (int k0 = 0; k0 < HH; k0 += 16) {
    if (k0 + 16 < HH) {
      issue_b(k0 + 16, buf ^ 1);   // async prefetch B (2 ops)
      stage_a(k0 + 16, buf ^ 1);   // VGPR-path prefetch A (BN+ReLU fused)
      WAIT_ASYNC(2);
    } else {
      WAIT_ASYNC(0);
    }
    __syncthreads();

#pragma unroll
    for (int ks = 0; ks < 4; ++ks) {
      const int kf = ks * 4 + 2 * lh;
      const v2f a0 = *(const v2f*)&As[buf][wm * 32 + ll][kf];
      const v2f a1 = *(const v2f*)&As[buf][wm * 32 + 16 + ll][kf];
#pragma unroll
      for (int nt = 0; nt < 4; ++nt) {
        const int coln = wn * 64 + nt * 16 + ll;
        v2f b;
        b[0] = Bs[buf][kf][coln];
        b[1] = Bs[buf][kf + 1][coln];
        acc[0][nt] = wmma_f32(a0, b, acc[0][nt]);
        acc[1][nt] = wmma_f32(a1, b, acc[1][nt]);
      }
    }
    __syncthreads();
    buf ^= 1;
  }

#pragma unroll
  for (int nt = 0; nt < 4; ++nt) {
    const int colg = obase + wn * 64 + nt * 16 + ll;
    const float bias = b2[(size_t)e * OO + colg];
#pragma unroll
    for (int mt = 0; mt < 2; ++mt)
#pragma unroll
      for (int j = 0; j < 8; ++j) {
        const int row = bbase + wm * 32 + mt * 16 + j + 8 * lh;
        out[((size_t)row * CC + c) * OO + colg] = acc[mt][nt][j] + bias;
      }
  }
}

// ---------------------------------------------------------------------------
// Host-side launch
// Inputs: 0:x 1:mask_ids 2:projection_idx 3:W1 4:b1 5:gamma 6:beta 7:W2 8:b2
// Output: out [B,C,O] f32 followed by mask_ids [B,C] (bit-copied).
// ---------------------------------------------------------------------------
extern "C" void kernel_launch(void* const* d_in, const int* in_sizes, int n_in,
                              void* d_out, int out_size, void* d_ws, size_t ws_size,
                              hipStream_t stream) {
  const float* x     = (const float*)d_in[0];
  const int*   mids  = (const int*)  d_in[1];
  const int*   pidx  = (const int*)  d_in[2];
  const float* W1    = (const float*)d_in[3];
  const float* b1    = (const float*)d_in[4];
  const float* gamma = (const float*)d_in[5];
  const float* beta  = (const float*)d_in[6];
  const float* W2    = (const float*)d_in[7];
  const float* b2    = (const float*)d_in[8];
  float* out = (float*)d_out;

  // Workspace layout
  char* ws = (char*)d_ws;
  float* h     = (float*)ws; ws += (size_t)BB * CC * HH * sizeof(float);
  float* part1 = (float*)ws; ws += (size_t)2 * CC * HH * sizeof(float);
  float* part2 = (float*)ws; ws += (size_t)2 * CC * HH * sizeof(float);
  float* s1    = (float*)ws; ws += (size_t)EE * HH * sizeof(float);
  float* s2    = (float*)ws; ws += (size_t)EE * HH * sizeof(float);
  float* scale = (float*)ws; ws += (size_t)EE * HH * sizeof(float);
  float* shift = (float*)ws; ws += (size_t)EE * HH * sizeof(float);

  moe_gemm1_kernel<<<dim3(HH / 128, BB / 128, CC), 256, 0, stream>>>(
      x, W1, b1, pidx, h, part1, part2);
  moe_stats_kernel<<<(EE * HH) / 256, 256, 0, stream>>>(part1, part2, pidx, s1, s2);
  moe_bnparam_kernel<<<(EE * HH) / 256, 256, 0, stream>>>(s1, s2, pidx, gamma, beta,
                                                          scale, shift);
  moe_gemm2_kernel<<<dim3(OO / 128, BB / 128, CC), 256, 0, stream>>>(
      h, W2, b2, pidx, scale, shift, out);

  // Second tuple element: mask_ids passthrough (bit-copy int32 into tail).
  hipMemcpyAsync(out + (size_t)BB * CC * OO, mids,
                 (size_t)BB * CC * sizeof(int), hipMemcpyDeviceToDevice, stream);
}